// EGCNConv_89567247991617
// MI455X (gfx1250) — compile-verified
//
#include <hip/hip_runtime.h>
#include <stdint.h>

typedef __attribute__((ext_vector_type(2))) float v2f;
typedef __attribute__((ext_vector_type(8))) float v8f;

#define D_FEAT 128

// ---------------------------------------------------------------------------
// Kernel 1: xw = x @ W^T via V_WMMA_F32_16X16X4_F32 (full fp32 accuracy).
// grid.x = N/16 row tiles, 256 threads = 8 waves, wave w handles column tile w.
// A (16x4 f32): lanes 0-15 -> M=lane, K=0,1 ; lanes 16-31 -> M=lane-16, K=2,3
// B (4x16 f32): lanes 0-15 -> N=lane, K=0,1 ; lanes 16-31 -> N=lane-16, K=2,3
//   B[k][n] = W[n][k]  (so a float2 row-load of W gives the B operand directly)
// D (16x16 f32): VGPR r -> M = (lane>>4)*8 + r, N = lane&15
// ---------------------------------------------------------------------------
__global__ void gcn_gemm_xwT_wmma(const float* __restrict__ x,
                                  const float* __restrict__ W,
                                  float* __restrict__ xw) {
  const int lane  = threadIdx.x & 31;
  const int wave  = threadIdx.x >> 5;            // column tile 0..7
  const int mtile = blockIdx.x;                  // row tile
  const int arow  = mtile * 16 + (lane & 15);    // x row this lane feeds
  const int bcol  = wave * 16 + (lane & 15);     // output column = W row
  const int khalf = (lane >> 4) * 2;             // 0 or 2

  const float* xp = x + arow * D_FEAT + khalf;
  const float* wp = W + bcol * D_FEAT + khalf;

  v8f acc = {};
#pragma unroll
  for (int k = 0; k < D_FEAT; k += 4) {
    v2f a = *(const v2f*)(xp + k);
    v2f b = *(const v2f*)(wp + k);
    acc = __builtin_amdgcn_wmma_f32_16x16x4_f32(
        /*neg_a=*/false, a, /*neg_b=*/false, b,
        /*c_mod=*/(short)0, acc, /*reuse_a=*/false, /*reuse_b=*/false);
  }

  const int rbase = mtile * 16 + (lane >> 4) * 8;
  const int col   = wave * 16 + (lane & 15);
#pragma unroll
  for (int r = 0; r < 8; ++r)
    xw[(rbase + r) * D_FEAT + col] = acc[r];
}

// ---------------------------------------------------------------------------
// Kernel 2: deg[i] = 1.0f (self-loop contribution)
// ---------------------------------------------------------------------------
__global__ void gcn_deg_init(float* __restrict__ deg, int n) {
  int i = blockIdx.x * blockDim.x + threadIdx.x;
  if (i < n) deg[i] = 1.0f;
}

// ---------------------------------------------------------------------------
// Kernel 3: deg[dst[e]] += 1 for every edge
// ---------------------------------------------------------------------------
__global__ void gcn_deg_count(const int* __restrict__ dst,
                              float* __restrict__ deg, int nEdges) {
  int e = blockIdx.x * blockDim.x + threadIdx.x;
  if (e < nEdges) atomicAdd(&deg[dst[e]], 1.0f);
}

// ---------------------------------------------------------------------------
// Kernel 4: dinv[i] = rsqrt(deg[i]) in place (deg >= 1 always)
// ---------------------------------------------------------------------------
__global__ void gcn_deg_rsqrt(float* __restrict__ deg, int n) {
  int i = blockIdx.x * blockDim.x + threadIdx.x;
  if (i < n) deg[i] = rsqrtf(deg[i]);
}

// ---------------------------------------------------------------------------
// Kernel 5: out = xw * dinv^2 (self-loop message) + bias
// ---------------------------------------------------------------------------
__global__ void gcn_self_bias(const float* __restrict__ xw,
                              const float* __restrict__ dinv,
                              const float* __restrict__ bias,
                              float* __restrict__ out, int n) {
  int t = blockIdx.x * blockDim.x + threadIdx.x;
  if (t >= n * D_FEAT) return;
  int i = t >> 7;          // node
  int c = t & (D_FEAT - 1);
  float dv = dinv[i];
  out[t] = xw[t] * dv * dv + bias[c];
}

// ---------------------------------------------------------------------------
// Kernel 6: scatter-add  out[dst] += xw[src] * (dinv[src]*dinv[dst])
// One wave per edge; lane owns 4 consecutive channels (float4 gather).
// out + xw both fit in the 192 MB L2, so atomics resolve at L2 rate.
// ---------------------------------------------------------------------------
__global__ void gcn_edge_scatter(const float* __restrict__ xw,
                                 const float* __restrict__ dinv,
                                 const int* __restrict__ src,
                                 const int* __restrict__ dst,
                                 float* __restrict__ out, int nEdges) {
  unsigned t = blockIdx.x * blockDim.x + threadIdx.x;
  int e = t >> 5;
  if (e >= nEdges) return;
  int lane = t & 31;
  int s = src[e];
  int d = dst[e];
  float norm = dinv[s] * dinv[d];
  const float4 v = *(const float4*)(xw + s * D_FEAT + lane * 4);
  float* o = out + d * D_FEAT + lane * 4;
  atomicAdd(o + 0, v.x * norm);
  atomicAdd(o + 1, v.y * norm);
  atomicAdd(o + 2, v.z * norm);
  atomicAdd(o + 3, v.w * norm);
}

// ---------------------------------------------------------------------------
// Launcher
// inputs: 0=x [N,128] f32, 1=W [128,128] f32, 2=b [128] f32,
//         3=edge_index [2,E] int32 (row 0 = src, row 1 = dst)
// ---------------------------------------------------------------------------
extern "C" void kernel_launch(void* const* d_in, const int* in_sizes, int n_in,
                              void* d_out, int out_size, void* d_ws, size_t ws_size,
                              hipStream_t stream) {
  const float* x = (const float*)d_in[0];
  const float* W = (const float*)d_in[1];
  const float* b = (const float*)d_in[2];
  const int* ei  = (const int*)d_in[3];

  const int nNodes = in_sizes[0] / D_FEAT;
  const int nEdges = in_sizes[3] / 2;
  const int* src = ei;
  const int* dst = ei + nEdges;

  float* out = (float*)d_out;

  // workspace: xw [nNodes*128] f32, then deg/dinv [nNodes] f32 (in place)
  float* xw  = (float*)d_ws;
  float* deg = (float*)((char*)d_ws + (size_t)nNodes * D_FEAT * sizeof(float));

  // 1) GEMM: one 16-row tile per block, 8 waves cover the 8 column tiles
  const int mtiles = (nNodes + 15) / 16;  // N = 100000 -> 6250 exact
  gcn_gemm_xwT_wmma<<<mtiles, 256, 0, stream>>>(x, W, xw);

  // 2) degree = 1 (self loop)
  gcn_deg_init<<<(nNodes + 255) / 256, 256, 0, stream>>>(deg, nNodes);

  // 3) += 1 per incoming edge
  gcn_deg_count<<<(nEdges + 255) / 256, 256, 0, stream>>>(dst, deg, nEdges);

  // 4) dinv = rsqrt(deg), in place
  gcn_deg_rsqrt<<<(nNodes + 255) / 256, 256, 0, stream>>>(deg, nNodes);

  // 5) out = self-loop message + bias
  {
    long long tot = (long long)nNodes * D_FEAT;
    int blocks = (int)((tot + 255) / 256);
    gcn_self_bias<<<blocks, 256, 0, stream>>>(xw, deg, b, out, nNodes);
  }

  // 6) edge scatter-add (one wave per edge)
  {
    long long tot = (long long)nEdges * 32;
    int blocks = (int)((tot + 255) / 256);
    gcn_edge_scatter<<<blocks, 256, 0, stream>>>(xw, deg, src, dst, out, nEdges);
  }
}